// Tip3pLike_50663434224255
// MI455X (gfx1250) — compile-verified
//
#include <hip/hip_runtime.h>
#include <hip/hip_bf16.h>

typedef __attribute__((ext_vector_type(2))) float v2f;
typedef __attribute__((ext_vector_type(8))) float v8f;

#define S_SITES   1536
#define N_TILES   96          // S_SITES / 16
#define N_FRAMES  16
#define COULOMB_K 332.0636f

// One block = one (frame, 16-row i-tile). Symmetric half-matrix schedule:
// row r handles tiles jt = (r+t) % 96 for t = 0..47 (plus t=48 iff r<48).
// t=0 is the diagonal tile (weight 1); t>0 tiles are weighted x2 (covers
// the transposed tile by symmetry). Every unordered tile pair is computed
// exactly once -> total work halved vs the full matrix.
//
// d^2 tile via a single V_WMMA_F32_16X16X4_F32:
//   A row m = [-2x_i, -2y_i, -2z_i, 1],  B col n = [x_j, y_j, z_j, |x_j|^2],
//   C[m][n] = |x_i|^2   =>   D = d^2.
__global__ __launch_bounds__(256) void tip3p_wmma_kernel(
    const float* __restrict__ pos,      // [B, S, 3]
    const float* __restrict__ charges,  // [B, S, 1]
    const float* __restrict__ ljp,      // [B, S, 2]
    float* __restrict__ partial)        // [B * N_TILES]
{
    __shared__ float xj[S_SITES], yj[S_SITES], zj[S_SITES], sqj[S_SITES];
    __shared__ float qj[S_SITES], sej[S_SITES], sgj[S_SITES];
    __shared__ float red[256];

    const int tid   = threadIdx.x;
    const int frame = blockIdx.x / N_TILES;
    const int row   = blockIdx.x % N_TILES;
    const int ibase = row * 16;

    const float* fpos = pos     + (size_t)frame * S_SITES * 3;
    const float* fq   = charges + (size_t)frame * S_SITES;
    const float* flj  = ljp     + (size_t)frame * S_SITES * 2;

    // ---- stage whole frame into LDS (precompute |x|^2 and sqrt(eps)) ----
    for (int s = tid; s < S_SITES; s += 256) {
        float x = fpos[s * 3 + 0];
        float y = fpos[s * 3 + 1];
        float z = fpos[s * 3 + 2];
        xj[s] = x; yj[s] = y; zj[s] = z;
        sqj[s] = x * x + y * y + z * z;
        qj[s]  = fq[s];
        sej[s] = __builtin_amdgcn_sqrtf(flj[s * 2 + 0]);
        sgj[s] = flj[s * 2 + 1];
    }
    __syncthreads();

    const int lane = tid & 31;
    const int half = lane >> 4;     // K/M half-wave split
    const int nloc = lane & 15;
    const int wave = tid >> 5;      // 0..7

    // ---- A fragment (16x4 f32): lanes 0-15 hold K=0,1 ; lanes 16-31 K=2,3 ----
    v2f A;
    {
        int i = ibase + nloc;
        if (half == 0) { A.x = -2.0f * xj[i]; A.y = -2.0f * yj[i]; }
        else           { A.x = -2.0f * zj[i]; A.y = 1.0f; }
    }

    // ---- per-accumulator-row i-side constants: m = k + 8*half ----
    float ci[8], qi[8], sei[8], sgi[8];
    int   imol[8];
    #pragma unroll
    for (int k = 0; k < 8; ++k) {
        int i   = ibase + k + 8 * half;
        ci[k]   = sqj[i];
        qi[k]   = COULOMB_K * qj[i];   // fold Coulomb constant into q_i
        sei[k]  = sej[i];
        sgi[k]  = sgj[i];
        imol[k] = i / 3;
    }

    // half-matrix task count for this row
    const int ntiles = 48 + (row < 48 ? 1 : 0);

    // task loader: t -> j-side fragment + constants (+ tile weight)
    auto load_task = [&](int t, v2f& Bm, float& qv, float& sev, float& sgv,
                         int& jm, float& wg) {
        int jt = row + t; if (jt >= N_TILES) jt -= N_TILES;
        int j  = jt * 16 + nloc;
        if (half == 0) { Bm.x = xj[j]; Bm.y = yj[j];  }
        else           { Bm.x = zj[j]; Bm.y = sqj[j]; }
        qv = qj[j]; sev = sej[j]; sgv = sgj[j];
        jm = j / 3;
        wg = (t == 0) ? 1.0f : 2.0f;
    };

    float acc = 0.0f;

    // ---- software-pipelined tile loop (prefetch next B while computing) ----
    int  t   = wave;
    bool has = (t < ntiles);
    v2f Bc; float qc, sec, sgc, wc; int jmc;
    if (has) load_task(t, Bc, qc, sec, sgc, jmc, wc);

    while (has) {
        const int  tn   = t + 8;
        const bool hasn = (tn < ntiles);
        v2f Bn; float qn, sen, sgn, wn; int jmn;
        if (hasn) load_task(tn, Bn, qn, sen, sgn, jmn, wn);

        v8f C;
        #pragma unroll
        for (int k = 0; k < 8; ++k) C[k] = ci[k];

        // D = d^2 tile (8 args: neg_a, A, neg_b, B, c_mod, C, reuse_a, reuse_b)
        v8f D = __builtin_amdgcn_wmma_f32_16x16x4_f32(
            false, A, false, Bc, (short)0, C, false, false);

        float tacc = 0.0f;
        #pragma unroll
        for (int k = 0; k < 8; ++k) {
            float d2    = fmaxf(D[k], 1e-12f);
            float inv_r = __builtin_amdgcn_rsqf(d2);
            float coul  = qi[k] * qc * inv_r;          // K already folded in qi
            float sig   = 0.5f * (sgi[k] + sgc);
            float s2    = (sig * sig) * (inv_r * inv_r);
            float s6    = s2 * s2 * s2;
            float lj    = 4.0f * sei[k] * sec * (s6 * s6 - s6);
            tacc += (imol[k] != jmc) ? (coul + lj) : 0.0f;
        }
        acc += wc * tacc;

        Bc = Bn; qc = qn; sec = sen; sgc = sgn; jmc = jmn; wc = wn;
        t = tn; has = hasn;
    }

    // ---- deterministic block reduction ----
    red[tid] = acc;
    __syncthreads();
    for (int off = 128; off > 0; off >>= 1) {
        if (tid < off) red[tid] += red[tid + off];
        __syncthreads();
    }
    if (tid == 0) partial[blockIdx.x] = red[0];
}

// Reduce 96 row-partials per frame into the final [B,1] energies.
__global__ __launch_bounds__(128) void tip3p_reduce_kernel(
    const float* __restrict__ partial, float* __restrict__ out)
{
    __shared__ float red[128];
    const int f   = blockIdx.x;
    const int tid = threadIdx.x;
    red[tid] = (tid < N_TILES) ? partial[f * N_TILES + tid] : 0.0f;
    __syncthreads();
    for (int off = 64; off > 0; off >>= 1) {
        if (tid < off) red[tid] += red[tid + off];
        __syncthreads();
    }
    if (tid == 0) out[f] = red[0];
}

extern "C" void kernel_launch(void* const* d_in, const int* in_sizes, int n_in,
                              void* d_out, int out_size, void* d_ws, size_t ws_size,
                              hipStream_t stream)
{
    const float* pos     = (const float*)d_in[0];  // [16,1536,3]
    const float* charges = (const float*)d_in[1];  // [16,1536,1]
    const float* ljp     = (const float*)d_in[2];  // [16,1536,2]
    float* out     = (float*)d_out;                // [16,1]
    float* partial = (float*)d_ws;                 // 16*96 floats = 6 KB scratch

    tip3p_wmma_kernel<<<N_FRAMES * N_TILES, 256, 0, stream>>>(pos, charges, ljp, partial);
    tip3p_reduce_kernel<<<N_FRAMES, 128, 0, stream>>>(partial, out);
}